// WeightedRuleLayer_56891136802881
// MI455X (gfx1250) — compile-verified
//
#include <hip/hip_runtime.h>
#include <hip/hip_bf16.h>

// CDNA5 / gfx1250. wave32. One wave computes a 16-neuron x 16-output tile:
//   D[16x16] += A[16x4] * B[4x16]  chained 32x over K=128 (8 slots * d_in 16)
// using v_wmma_f32_16x16x4_f32 (full fp32 precision; kernel is gather-bound,
// source_values (64MB) is L2-resident on MI455X's 192MB L2).
//
// All gather/store addresses are formed as SGPR-base + zext(32-bit VGPR byte
// offset) so the backend emits GVS-form global_load/store (saddr + voffset +
// immediate), minimizing per-access VALU address math.

typedef __attribute__((ext_vector_type(2))) float v2f;
typedef __attribute__((ext_vector_type(8))) float v8f;

#define INPUTS_DIM 8
#define D_IN 16
#define D_OUT 16

__global__ __launch_bounds__(256)
void weighted_rule_wmma_kernel(const float* __restrict__ src,     // [N_SOURCE, 16]
                               const float* __restrict__ wt,      // [8, 16, 16] row-major
                               const int*   __restrict__ idx,     // [n_neurons, 8]
                               float*       __restrict__ out,     // [n_neurons, 16]
                               int n_neurons, int n_tiles) {
    const int lane   = threadIdx.x & 31;
    const int m      = lane & 15;        // row-within-tile / output column
    const int hi     = lane >> 4;        // 0: lanes 0-15, 1: lanes 16-31
    const int wib    = threadIdx.x >> 5; // wave index in block
    const int wpb    = blockDim.x >> 5;  // waves per block
    const int wave   = blockIdx.x * wpb + wib;
    const int nwaves = gridDim.x * wpb;

    const char* srcB = reinterpret_cast<const char*>(src);
    const char* idxB = reinterpret_cast<const char*>(idx);
    char*       outB = reinterpret_cast<char*>(out);

    // ---- Preload all 32 B fragments (stacked weight, K=128 x N=16) --------
    // B layout for V_WMMA_F32_16X16X4_F32 (4x16, K x N):
    //   lane (n = lane&15, hi): VGPR0 = W[4kk+2hi][n], VGPR1 = W[4kk+2hi+1][n]
    // Flat weight index: W[k][n] = wt[k*16 + n] with k = slot*16 + d.
    v2f bfrag[32];
#pragma unroll
    for (int kk = 0; kk < 32; ++kk) {
        const int k0 = 4 * kk + 2 * hi;
        v2f b;
        b[0] = wt[k0 * D_OUT + m];
        b[1] = wt[(k0 + 1) * D_OUT + m];
        bfrag[kk] = b;
    }

    // ---- Grid-stride over 16-neuron tiles ---------------------------------
    for (int tile = wave; tile < n_tiles; tile += nwaves) {
        const int nbase = tile * 16;
        int row = nbase + m;
        if (row > n_neurons - 1) row = n_neurons - 1;   // defensive clamp

        // Load this row's 8 gather indices (two int4 = 32B, aligned).
        // 32-bit byte offset (row < 2^26) -> GVS-form global_load_b128.
        const unsigned rowOff = (unsigned)row * (INPUTS_DIM * 4u);
        const int4 i0 = *reinterpret_cast<const int4*>(idxB + (size_t)rowOff);
        const int4 i1 = *reinterpret_cast<const int4*>(idxB + (size_t)(rowOff + 16u));
        int sidx[INPUTS_DIM];
        sidx[0] = i0.x; sidx[1] = i0.y; sidx[2] = i0.z; sidx[3] = i0.w;
        sidx[4] = i1.x; sidx[5] = i1.y; sidx[6] = i1.z; sidx[7] = i1.w;

        // Per-slot 32-bit base byte offset into source: idx*64 + hi*8.
        // The (kk&3)*16 part folds into the load's immediate offset.
        unsigned soff[INPUTS_DIM];
#pragma unroll
        for (int s = 0; s < INPUTS_DIM; ++s)
            soff[s] = ((unsigned)sidx[s] << 6) + ((unsigned)hi << 3);

        v8f c = {};
#pragma unroll
        for (int kk = 0; kk < 32; ++kk) {
            const int slot = kk >> 2;                 // K slice 4*kk lives in slot kk/4
            const unsigned off = soff[slot] + (unsigned)((kk & 3) << 4);
            v2f a = *reinterpret_cast<const v2f*>(srcB + (size_t)off); // global_load_b64 saddr-form
            // (neg_a, A, neg_b, B, c_mod, C, reuse_a, reuse_b)
            c = __builtin_amdgcn_wmma_f32_16x16x4_f32(
                    false, a, false, bfrag[kk], (short)0, c, false, false);
        }

        // ---- tanh + store -------------------------------------------------
        // C/D layout: VGPR v -> M = v + 8*hi, N = lane&15.
        const int Mbase = nbase + hi * 8;
        const unsigned obase = (unsigned)Mbase * (D_OUT * 4u) + ((unsigned)m << 2);
        if (nbase + 16 <= n_neurons) {
#pragma unroll
            for (int v = 0; v < 8; ++v) {
                *reinterpret_cast<float*>(outB + (size_t)(obase + (unsigned)v * (D_OUT * 4u)))
                    = tanhf(c[v]);
            }
        } else {
#pragma unroll
            for (int v = 0; v < 8; ++v) {
                if (Mbase + v < n_neurons)
                    *reinterpret_cast<float*>(outB + (size_t)(obase + (unsigned)v * (D_OUT * 4u)))
                        = tanhf(c[v]);
            }
        }
    }
}

extern "C" void kernel_launch(void* const* d_in, const int* in_sizes, int n_in,
                              void* d_out, int out_size, void* d_ws, size_t ws_size,
                              hipStream_t stream) {
    const float* src = (const float*)d_in[0];   // source_values [N_SOURCE*16] fp32
    const float* wt  = (const float*)d_in[1];   // weight        [8*16*16]    fp32
    const int*   idx = (const int*)d_in[2];     // gather_indices [N*8] (int32 under JAX x64-off)
    float*       out = (float*)d_out;           // [N*16] fp32

    const int n_neurons = in_sizes[2] / INPUTS_DIM;
    const int n_tiles   = (n_neurons + 15) / 16;

    const int threads = 256;                    // 8 waves per block
    const int wpb     = threads / 32;
    int grid = (n_tiles + wpb - 1) / wpb;
    if (grid > 2048) grid = 2048;               // grid-stride: reuse preloaded B frags
    if (grid < 1) grid = 1;

    weighted_rule_wmma_kernel<<<grid, threads, 0, stream>>>(
        src, wt, idx, out, n_neurons, n_tiles);
}